// DTM_14757507629484
// MI455X (gfx1250) — compile-verified
//
#include <hip/hip_runtime.h>
#include <hip/hip_bf16.h>
#include <math.h>

#define MEM_LEN   100000
#define ZD        256
#define K_NN      7
#define NUM_EDGES 500000
#define M_TILES   (MEM_LEN / 16)   // 6250

typedef float v2f __attribute__((ext_vector_type(2)));
typedef float v8f __attribute__((ext_vector_type(8)));

// ---------------------------------------------------------------------------
// 1) d2[m] = || mems2[m] - r ||^2   (row y substituted by r -> exact 0)
//    one wave (32 lanes) per row, strided coalesced loads
// ---------------------------------------------------------------------------
__global__ void k_d2(const float* __restrict__ mems, const float* __restrict__ r,
                     const int* __restrict__ y, float* __restrict__ d2) {
    int row  = blockIdx.x * 8 + (threadIdx.x >> 5);
    int lane = threadIdx.x & 31;
    if (row >= MEM_LEN) return;
    int yv = y[0];
    float s = 0.f;
#pragma unroll
    for (int i = 0; i < 8; ++i) {
        int k = i * 32 + lane;
        float rv = r[k];
        float x  = (row == yv) ? rv : mems[row * ZD + k];
        float d  = x - rv;
        s += d * d;
    }
    for (int off = 16; off; off >>= 1) s += __shfl_down(s, off, 32);
    if (lane == 0) d2[row] = s;
}

// ---------------------------------------------------------------------------
// 2) top-K smallest d2 (K=7), ties -> lowest index (jax top_k stability).
//    single block, 7 sequential global arg-min passes; d2 lives in L2 (400KB)
// ---------------------------------------------------------------------------
__global__ void k_topk(float* __restrict__ d2, int* __restrict__ nbr) {
    __shared__ float sv[1024];
    __shared__ int   si[1024];
    int tid = threadIdx.x;
    for (int iter = 0; iter < K_NN; ++iter) {
        float bv = 3.4e38f; int bi = 0x7fffffff;
        for (int i = tid; i < MEM_LEN; i += 1024) {
            float v = d2[i];
            if (v < bv || (v == bv && i < bi)) { bv = v; bi = i; }
        }
        sv[tid] = bv; si[tid] = bi;
        __syncthreads();
        for (int s = 512; s; s >>= 1) {
            if (tid < s) {
                float ov = sv[tid + s]; int oi = si[tid + s];
                if (ov < sv[tid] || (ov == sv[tid] && oi < si[tid])) {
                    sv[tid] = ov; si[tid] = oi;
                }
            }
            __syncthreads();
        }
        if (tid == 0) { nbr[iter] = si[0]; d2[si[0]] = 3.4e38f; }
        __syncthreads();
    }
}

// ---------------------------------------------------------------------------
// 3) deg init: 2.0 everywhere (improved self loop), + K at row y
//    (the K kNN edges all target y with weight 1; edges targeting y are masked)
// ---------------------------------------------------------------------------
__global__ void k_init_deg(float* __restrict__ deg, const int* __restrict__ y) {
    int i = blockIdx.x * 256 + threadIdx.x;
    if (i < MEM_LEN) deg[i] = 2.0f + ((i == y[0]) ? (float)K_NN : 0.0f);
}

// 4) deg += 1 per surviving edge (exact small-int f32 atomic sums -> deterministic)
__global__ void k_edge_deg(const int* __restrict__ ei, const int* __restrict__ y,
                           float* __restrict__ deg) {
    int e = blockIdx.x * 256 + threadIdx.x;
    if (e >= NUM_EDGES) return;
    int tgt = ei[NUM_EDGES + e];
    if (tgt != y[0]) atomicAdd(&deg[tgt], 1.0f);
}

// 5) dinv = 1/sqrt(deg)
__global__ void k_dinv(const float* __restrict__ deg, float* __restrict__ dinv) {
    int i = blockIdx.x * 256 + threadIdx.x;
    if (i < MEM_LEN) dinv[i] = 1.0f / sqrtf(deg[i]);
}

// ---------------------------------------------------------------------------
// 6) xw = mems2 @ W  via V_WMMA_F32_16X16X4_F32 (exact f32 accumulate).
//    One 16x16 output tile per wave; 8 waves (tiles) per 256-thread block.
//    A 16x4 layout: lanes 0-15 hold M=0..15, VGPR0/1 = K {0,1}; lanes 16-31 K {2,3}.
//    B 4x16 mirrored; C/D: VGPR j -> M = j + 8*(lane>=16), N = lane&15.
// ---------------------------------------------------------------------------
__global__ void k_gemm(const float* __restrict__ mems, const float* __restrict__ r,
                       const float* __restrict__ W, const int* __restrict__ y,
                       float* __restrict__ xw) {
    int lane   = threadIdx.x & 31;
    int wave   = threadIdx.x >> 5;
    int tile_m = blockIdx.y * 8 + wave;
    if (tile_m >= M_TILES) return;
    int nbase = blockIdx.x * 16;
    int mbase = tile_m * 16;
    int yv    = y[0];
    int half  = lane >> 4;      // 0 or 1
    int l15   = lane & 15;
    int mrow  = mbase + l15;    // A-matrix row held by this lane
    int ncol  = nbase + l15;    // B-matrix column held by this lane
    bool isY  = (mrow == yv);

    v8f c = {};
    for (int k = 0; k < ZD; k += 4) {
        int k0 = k + 2 * half;
        v2f a, b;
        a.x = isY ? r[k0]     : mems[mrow * ZD + k0];
        a.y = isY ? r[k0 + 1] : mems[mrow * ZD + k0 + 1];
        b.x = W[k0 * ZD + ncol];
        b.y = W[(k0 + 1) * ZD + ncol];
        c = __builtin_amdgcn_wmma_f32_16x16x4_f32(
                /*neg_a=*/false, a, /*neg_b=*/false, b,
                /*c_mod=*/(short)0, c, /*reuse_a=*/false, /*reuse_b=*/false);
    }

    int mout = mbase + 8 * half;
#pragma unroll
    for (int j = 0; j < 8; ++j)
        xw[(mout + j) * ZD + ncol] = c[j];
}

// ---------------------------------------------------------------------------
// 7) out = (2 * dinv^2) * xw   (initializes d_out; runs before edge atomics)
// ---------------------------------------------------------------------------
__global__ void k_selfloop(const float* __restrict__ xw, const float* __restrict__ dinv,
                           float* __restrict__ out) {
    int i = blockIdx.x * 256 + threadIdx.x;   // < 25,600,000 fits int
    int m = i / ZD;
    float di = dinv[m];
    out[i] = 2.0f * di * di * xw[i];
}

// ---------------------------------------------------------------------------
// 8) edge message pass: out[tgt] += dinv[src]*dinv[tgt] * xw[src]
//    one wave per edge; extra K virtual edges (nbr -> y). BW-dominant step.
// ---------------------------------------------------------------------------
__global__ void k_edge_msg(const int* __restrict__ ei, const int* __restrict__ y,
                           const int* __restrict__ nbr, const float* __restrict__ dinv,
                           const float* __restrict__ xw, float* __restrict__ out) {
    int e    = blockIdx.x * 8 + (threadIdx.x >> 5);
    int lane = threadIdx.x & 31;
    if (e >= NUM_EDGES + K_NN) return;
    int yv = y[0];
    int src, tgt;
    if (e < NUM_EDGES) {
        src = ei[e];
        tgt = ei[NUM_EDGES + e];
        if (tgt == yv) return;            // masked edge (weight 0)
    } else {
        src = nbr[e - NUM_EDGES];
        tgt = yv;                          // kNN edge, weight 1
    }
    float norm = dinv[src] * dinv[tgt];
    const float* xs = xw  + src * ZD;
    float*       od = out + tgt * ZD;
#pragma unroll
    for (int i = 0; i < 8; ++i) {
        int z = i * 32 + lane;
        atomicAdd(&od[z], norm * xs[z]);
    }
}

extern "C" void kernel_launch(void* const* d_in, const int* in_sizes, int n_in,
                              void* d_out, int out_size, void* d_ws, size_t ws_size,
                              hipStream_t stream) {
    const float* mems = (const float*)d_in[0];
    const float* r    = (const float*)d_in[1];
    const float* W    = (const float*)d_in[2];
    const int*   y    = (const int*)d_in[3];
    const int*   ei   = (const int*)d_in[4];
    float* out = (float*)d_out;

    // workspace layout (all 16B aligned): xw | d2 | deg | dinv | nbr
    float* xw   = (float*)d_ws;
    float* d2   = xw   + (size_t)MEM_LEN * ZD;   // 25,600,000 floats
    float* deg  = d2   + MEM_LEN;
    float* dinv = deg  + MEM_LEN;
    int*   nbr  = (int*)(dinv + MEM_LEN);

    k_d2      <<<(MEM_LEN + 7) / 8,        256, 0, stream>>>(mems, r, y, d2);
    k_topk    <<<1,                        1024, 0, stream>>>(d2, nbr);
    k_init_deg<<<(MEM_LEN + 255) / 256,    256, 0, stream>>>(deg, y);
    k_edge_deg<<<(NUM_EDGES + 255) / 256,  256, 0, stream>>>(ei, y, deg);
    k_dinv    <<<(MEM_LEN + 255) / 256,    256, 0, stream>>>(deg, dinv);

    dim3 gg(ZD / 16, (M_TILES + 7) / 8);   // 16 x 782
    k_gemm    <<<gg,                       256, 0, stream>>>(mems, r, W, y, xw);

    k_selfloop<<<(MEM_LEN * ZD) / 256,     256, 0, stream>>>(xw, dinv, out);
    k_edge_msg<<<(NUM_EDGES + K_NN + 7)/8, 256, 0, stream>>>(ei, y, nbr, dinv, xw, out);
}